// SO8TAttention_52484500357705
// MI455X (gfx1250) — compile-verified
//
#include <hip/hip_runtime.h>
#include <hip/hip_bf16.h>

#define N_HEADS 32
#define N_KV 8
#define HDIM 8
#define SEQ 2048
#define HID 256
#define QTILES (SEQ / 16)

typedef __attribute__((ext_vector_type(2))) float v2f;
typedef __attribute__((ext_vector_type(8))) float v8f;

// ---------------------------------------------------------------------------
// 16-lane max reduction entirely in DPP16 VALU ops (no LDS, no waits).
// Butterfly: xor1 (quad_perm 1,0,3,2), xor2 (quad_perm 2,3,0,1),
// xor7-in-8 (row_half_mirror), xor15-in-16 (row_mirror). Never crosses the
// 16-lane half, so both WMMA half-rows reduce independently.
// ---------------------------------------------------------------------------
template <int CTRL>
__device__ __forceinline__ float dpp_max(float x) {
  const int y =
      __builtin_amdgcn_update_dpp(0, __float_as_int(x), CTRL, 0xf, 0xf, true);
  return fmaxf(x, __int_as_float(y));
}

__device__ __forceinline__ float rowmax16(float x) {
  x = dpp_max<0xB1>(x);   // quad_perm(1,0,3,2)
  x = dpp_max<0x4E>(x);   // quad_perm(2,3,0,1)
  x = dpp_max<0x141>(x);  // row_half_mirror
  x = dpp_max<0x140>(x);  // row_mirror
  return x;
}

// Raw v_exp_f32: exact exp2 for x <= 0; underflows to 0 (what softmax wants).
__device__ __forceinline__ float fast_exp2(float x) {
  return __builtin_amdgcn_exp2f(x);
}

// ---------------------------------------------------------------------------
// Kernel 1: fused QKV projection + SO8T rotary + SO(8) rotation.
// One block per (b,s) row; 64 threads; threads 0..31 -> q heads,
// 32..39 -> k heads, 40..47 -> v heads. v is stored transposed [B,KV,8,S].
// ---------------------------------------------------------------------------
__global__ __launch_bounds__(64) void qkv_rope_so8_kernel(
    const float* __restrict__ x, const float* __restrict__ Wq,
    const float* __restrict__ Wk, const float* __restrict__ Wv,
    const float* __restrict__ rope, const float* __restrict__ so8,
    float* __restrict__ qb, float* __restrict__ kb, float* __restrict__ vtb) {
  const int bs = blockIdx.x;
  const int s = bs & (SEQ - 1);
  const int b = bs >> 11;

  __shared__ float xs[HID];
  for (int i = threadIdx.x; i < HID; i += 64) xs[i] = x[(size_t)bs * HID + i];
  __syncthreads();

  const int t = threadIdx.x;
  if (t >= 48) return;
  const int kind = (t < 32) ? 0 : (t < 40 ? 1 : 2);
  const int head = (t < 32) ? t : (t < 40 ? t - 32 : t - 40);
  const float* W = (kind == 0) ? Wq : ((kind == 1) ? Wk : Wv);
  const float* wr = W + (size_t)head * HDIM * HID;

  float acc[HDIM] = {};
  for (int i = 0; i < HID; ++i) {
    const float xv = xs[i];
#pragma unroll
    for (int d = 0; d < HDIM; ++d) acc[d] = fmaf(xv, wr[d * HID + i], acc[d]);
  }

  float y[HDIM];
  if (kind <= 1) {
    // rotated rope: cos/sin(freqs) @ rope_rot[:4,:4]
    const float invf[4] = {1.0f, 0.1f, 0.01f, 0.001f};
    float cr[4] = {}, sr[4] = {};
#pragma unroll
    for (int i = 0; i < 4; ++i) {
      const float fr = (float)s * invf[i];
      const float ci = cosf(fr), si = sinf(fr);
#pragma unroll
      for (int j = 0; j < 4; ++j) {
        cr[j] = fmaf(ci, rope[i * 8 + j], cr[j]);
        sr[j] = fmaf(si, rope[i * 8 + j], sr[j]);
      }
    }
#pragma unroll
    for (int d = 0; d < HDIM; ++d) {
      const int j = d & 3;
      const float rh = (d < 4) ? -acc[d + 4] : acc[d - 4];
      y[d] = acc[d] * cr[j] + rh * sr[j];
    }
  } else {
#pragma unroll
    for (int d = 0; d < HDIM; ++d) y[d] = acc[d];
  }

  // SO(8) rotation: z = y @ so8
  float z[HDIM];
#pragma unroll
  for (int j = 0; j < HDIM; ++j) {
    float sum = 0.0f;
#pragma unroll
    for (int d = 0; d < HDIM; ++d) sum = fmaf(y[d], so8[d * 8 + j], sum);
    z[j] = sum;
  }

  if (kind == 0) {
    float* dst = qb + (((size_t)b * N_HEADS + head) * SEQ + s) * HDIM;
#pragma unroll
    for (int j = 0; j < HDIM; ++j) dst[j] = z[j];
  } else if (kind == 1) {
    float* dst = kb + (((size_t)b * N_KV + head) * SEQ + s) * HDIM;
#pragma unroll
    for (int j = 0; j < HDIM; ++j) dst[j] = z[j];
  } else {
#pragma unroll
    for (int j = 0; j < HDIM; ++j)
      vtb[(((size_t)b * N_KV + head) * HDIM + j) * SEQ + s] = z[j];
  }
}

// ---------------------------------------------------------------------------
// One 16-key tile of flash attention. Q is pre-scaled by (1/sqrt(8))*log2(e)
// so softmax runs in the exp2 domain (identical result, no per-tile muls).
// MASK=true only for the peeled diagonal tile; the mask is a per-lane
// additive constant (key>row <=> lanelo > r+8*hi, independent of qt).
// ---------------------------------------------------------------------------
template <bool MASK>
__device__ __forceinline__ void attn_tile(
    const float* __restrict__ kp, const float* __restrict__ vp,
    float* __restrict__ pw, int kbase, int lanelo, int hi, int hi2,
    const float* __restrict__ maskv, const v2f Qa0, const v2f Qa1,
    float* __restrict__ m, v8f& o) {
  // K^T in B-layout: lane L, vgpr j holds K[kbase+(L&15)][4t+2*(L>>4)+j]
  const float* krow = kp + (size_t)(kbase + lanelo) * HDIM;
  const v2f Kb0 = *(const v2f*)(krow + hi2);
  const v2f Kb1 = *(const v2f*)(krow + 4 + hi2);

  v8f sc = {};
  sc = __builtin_amdgcn_wmma_f32_16x16x4_f32(false, Qa0, false, Kb0, (short)0,
                                             sc, false, false);
  sc = __builtin_amdgcn_wmma_f32_16x16x4_f32(false, Qa1, false, Kb1, (short)0,
                                             sc, false, false);

  float p[8];
#pragma unroll
  for (int r = 0; r < 8; ++r) {
    float sv = sc[r];
    if (MASK) sv += maskv[r];  // adds -3e38 on masked (key>row) elements
    const float rm = rowmax16(sv);
    const float mnew = fmaxf(m[r], rm);
    const float corr = fast_exp2(m[r] - mnew);
    m[r] = mnew;
    p[r] = fast_exp2(sv - mnew);
    o[r] = o[r] * corr;  // element-wise rescale (row index matches stats)
  }

  // P: D-layout -> A-layout via wave-private LDS (DS ops in-order per wave)
#pragma unroll
  for (int r = 0; r < 8; ++r) pw[r * 16 + hi * 128 + lanelo] = p[r];
  v2f Pa[4];
#pragma unroll
  for (int t = 0; t < 4; ++t)
    Pa[t] = *(const v2f*)(pw + (lanelo & 7) * 16 + (lanelo >> 3) * 128 +
                          4 * t + hi2);

  // V in B-layout from transposed v; column 8 = ones (denominator trick)
  v2f Vb[4];
#pragma unroll
  for (int t = 0; t < 4; ++t) {
    if (lanelo < 8) {
      Vb[t] = *(const v2f*)(vp + (size_t)lanelo * SEQ + kbase + 4 * t + hi2);
    } else {
      const float c = (lanelo == 8) ? 1.0f : 0.0f;
      Vb[t] = (v2f){c, c};
    }
  }
#pragma unroll
  for (int t = 0; t < 4; ++t)
    o = __builtin_amdgcn_wmma_f32_16x16x4_f32(false, Pa[t], false, Vb[t],
                                              (short)0, o, false, false);
}

// ---------------------------------------------------------------------------
// Kernel 2: flash-style causal attention, one wave32 per 16-row q tile.
// ---------------------------------------------------------------------------
__global__ __launch_bounds__(256) void so8_flash_attn_kernel(
    const float* __restrict__ q, const float* __restrict__ k,
    const float* __restrict__ vT, float* __restrict__ attn) {
  const int lane = threadIdx.x & 31;
  const int widx = threadIdx.x >> 5;
  const int w = blockIdx.x * 8 + widx;  // 0..8191
  const int qt = w & (QTILES - 1);
  const int h = (w >> 7) & (N_HEADS - 1);
  const int b = w >> 12;
  const int kv = h >> 2;  // repeat_interleave(4)
  const int qbase = qt * 16;
  const int lanelo = lane & 15;
  const int hi = lane >> 4;  // which 16-lane half
  const int hi2 = hi * 2;

  const float* qp = q + ((size_t)b * N_HEADS + h) * SEQ * HDIM;
  const float* kp = k + ((size_t)b * N_KV + kv) * SEQ * HDIM;
  const float* vp = vT + ((size_t)b * N_KV + kv) * HDIM * SEQ;

  __shared__ float plds[8][256];
  float* pw = &plds[widx][0];

  // Q in WMMA A-layout, pre-scaled into the exp2 softmax domain
  const float QSCALE = 0.35355339059327f * 1.44269504088896f;
  const float* qrow = qp + (size_t)(qbase + lanelo) * HDIM;
  const v2f Qa0 = *(const v2f*)(qrow + hi2) * QSCALE;
  const v2f Qa1 = *(const v2f*)(qrow + 4 + hi2) * QSCALE;

  // causal mask addend for the diagonal tile (static per lane/register)
  float maskv[8];
#pragma unroll
  for (int r = 0; r < 8; ++r)
    maskv[r] = (lanelo > r + 8 * hi) ? -3.0e38f : 0.0f;

  v8f o = {};  // D-layout: vgpr r, lane L -> (row r+8*hi, col lanelo)
  float m[8];
#pragma unroll
  for (int r = 0; r < 8; ++r) m[r] = -3.0e38f;

  // off-diagonal tiles: no masking at all
  for (int kt = 0; kt < qt; ++kt)
    attn_tile<false>(kp, vp, pw, kt * 16, lanelo, hi, hi2, maskv, Qa0, Qa1, m,
                     o);
  // peeled diagonal tile with the static mask addend
  attn_tile<true>(kp, vp, pw, qbase, lanelo, hi, hi2, maskv, Qa0, Qa1, m, o);

  // column 8 of the accumulator holds l (softmax denominator) per row
#pragma unroll
  for (int r = 0; r < 8; ++r) {
    const float ov = o[r];
    const float l = __shfl(ov, (lane & 16) + 8);
    const float outv = ov * __builtin_amdgcn_rcpf(l);  // l >= 1 always
    const int row = qbase + r + 8 * hi;
    if (lanelo < HDIM)
      attn[(((size_t)b * SEQ + row) * N_HEADS + h) * HDIM + lanelo] = outv;
  }
}

// ---------------------------------------------------------------------------
// Kernel 3: output projection  out = attn_flat @ Wo.T
// ---------------------------------------------------------------------------
__global__ __launch_bounds__(256) void out_proj_kernel(
    const float* __restrict__ attn, const float* __restrict__ Wo,
    float* __restrict__ out) {
  const int bs = blockIdx.x;
  __shared__ float xs[HID];
  xs[threadIdx.x] = attn[(size_t)bs * HID + threadIdx.x];
  __syncthreads();
  const int oi = threadIdx.x;
  const float* wr = Wo + (size_t)oi * HID;
  float acc = 0.0f;
  for (int i = 0; i < HID; ++i) acc = fmaf(xs[i], wr[i], acc);
  out[(size_t)bs * HID + oi] = acc;
}

// ---------------------------------------------------------------------------
extern "C" void kernel_launch(void* const* d_in, const int* in_sizes, int n_in,
                              void* d_out, int out_size, void* d_ws,
                              size_t ws_size, hipStream_t stream) {
  const float* x = (const float*)d_in[0];
  const float* Wq = (const float*)d_in[1];
  const float* Wk = (const float*)d_in[2];
  const float* Wv = (const float*)d_in[3];
  const float* Wo = (const float*)d_in[4];
  const float* rope = (const float*)d_in[5];
  const float* so8 = (const float*)d_in[6];

  float* ws = (float*)d_ws;
  float* qb = ws;                                     // 2*32*2048*8 = 1048576
  float* kb = qb + (size_t)2 * N_HEADS * SEQ * HDIM;  // 2*8*2048*8 = 262144
  float* vtb = kb + (size_t)2 * N_KV * SEQ * HDIM;    // 262144
  float* ab = vtb + (size_t)2 * N_KV * SEQ * HDIM;    // 1048576

  const int BS = 2 * SEQ;  // 4096 (b,s) rows

  qkv_rope_so8_kernel<<<BS, 64, 0, stream>>>(x, Wq, Wk, Wv, rope, so8, qb, kb,
                                             vtb);
  // 8192 q-tiles total, 8 waves (256 threads) per block
  so8_flash_attn_kernel<<<(2 * N_HEADS * QTILES) / 8, 256, 0, stream>>>(
      qb, kb, vtb, ab);
  out_proj_kernel<<<BS, 256, 0, stream>>>(ab, Wo, (float*)d_out);
}